// GradTreeBlock_62268435858112
// MI455X (gfx1250) — compile-verified
//
#include <hip/hip_runtime.h>

// Chain of 6 scalar affine+ReLU stages over 2^25 fp32 elements.
// Pure streaming problem: 256 MiB of HBM traffic, ~11.5 us floor at 23.3 TB/s.
// Memory-bound (AI ~2.25 flops/byte) -> WMMA deliberately unused.
// This revision: all-32-bit indexing (SADDR + 32-bit voffset addressing mode)
// and a uniform in-bounds fast path to strip 64-bit address math and
// per-element guards out of the hot loop.

#define DEPTH 6
#define THREADS 256
#define UNROLL 4  // float4 per thread per tile; tile = 256*4*16B = 16 KiB

typedef __attribute__((ext_vector_type(4))) float v4f;

__global__ __launch_bounds__(THREADS) void relu_chain6_kernel(
    const float* __restrict__ x, const float* __restrict__ w,
    const float* __restrict__ b, float* __restrict__ out, int n) {
  // Uniform scalar parameters -> SGPRs, live across the whole loop.
  float wr[DEPTH], br[DEPTH];
#pragma unroll
  for (int i = 0; i < DEPTH; ++i) {
    wr[i] = w[i];
    br[i] = b[i];
  }

  const int n4 = n >> 2;  // number of float4 elements (fits in int: 2^23)
  const v4f* __restrict__ x4 = (const v4f*)x;
  v4f* __restrict__ o4 = (v4f*)out;

  const int tileElems = THREADS * UNROLL;        // float4 per tile
  const int tileStride = (int)gridDim.x * tileElems;
  const int tid = (int)threadIdx.x;

  for (int tile = (int)blockIdx.x * tileElems; tile < n4; tile += tileStride) {
    // Prefetch the NEXT tile this block will process (gfx1250
    // global_prefetch_b8, no counter cost). One lane per 64 B cacheline:
    // 256 lanes * 64 B = 16 KiB = exactly one tile.
    const int pf = tile + tileStride + tid * UNROLL;
    if (pf < n4) {
      __builtin_prefetch((const void*)(x4 + pf), 0, 0);
    }

    if (tile + tileElems <= n4) {
      // Fast path: whole tile in bounds (uniform branch). No per-element
      // guards -> the 4 NT b128 loads issue as a clause.
      v4f v[UNROLL];
#pragma unroll
      for (int k = 0; k < UNROLL; ++k) {
        v[k] = __builtin_nontemporal_load(x4 + (tile + k * THREADS + tid));
      }
#pragma unroll
      for (int k = 0; k < UNROLL; ++k) {
#pragma unroll
        for (int i = 0; i < DEPTH; ++i) {
#pragma unroll
          for (int c = 0; c < 4; ++c) {
            v[k][c] = fmaxf(fmaf(wr[i], v[k][c], br[i]), 0.0f);
          }
        }
      }
#pragma unroll
      for (int k = 0; k < UNROLL; ++k) {
        __builtin_nontemporal_store(v[k], o4 + (tile + k * THREADS + tid));
      }
    } else {
      // Ragged last tile (not taken for N = 2^25; kept for general N).
#pragma unroll
      for (int k = 0; k < UNROLL; ++k) {
        const int j = tile + k * THREADS + tid;
        if (j < n4) {
          v4f v = __builtin_nontemporal_load(x4 + j);
#pragma unroll
          for (int i = 0; i < DEPTH; ++i) {
#pragma unroll
            for (int c = 0; c < 4; ++c) {
              v[c] = fmaxf(fmaf(wr[i], v[c], br[i]), 0.0f);
            }
          }
          __builtin_nontemporal_store(v, o4 + j);
        }
      }
    }
  }

  // Scalar tail (n % 4) — zero for N = 2^25, kept for correctness on any N.
  if (blockIdx.x == 0) {
    const int base = n4 << 2;
    const int rem = n - base;
    if ((int)threadIdx.x < rem) {
      float v = x[base + threadIdx.x];
#pragma unroll
      for (int i = 0; i < DEPTH; ++i) {
        v = fmaxf(fmaf(wr[i], v, br[i]), 0.0f);
      }
      out[base + threadIdx.x] = v;
    }
  }
}

extern "C" void kernel_launch(void* const* d_in, const int* in_sizes, int n_in,
                              void* d_out, int out_size, void* d_ws,
                              size_t ws_size, hipStream_t stream) {
  (void)n_in;
  (void)d_ws;
  (void)ws_size;
  (void)out_size;

  const float* x = (const float*)d_in[0];
  const float* w = (const float*)d_in[1];
  const float* b = (const float*)d_in[2];
  float* out = (float*)d_out;

  const int n = in_sizes[0];
  const int n4 = n >> 2;

  const int tileElems = THREADS * UNROLL;  // 1024 float4 = 16 KiB
  int tiles = (n4 + tileElems - 1) / tileElems;
  if (tiles < 1) tiles = 1;
  // Persistent-ish grid: saturate every WGP, but keep >1 tile per block so
  // the next-tile prefetch has work to do (8192 tiles / 2048 blocks = 4).
  const int grid = tiles < 2048 ? tiles : 2048;

  relu_chain6_kernel<<<grid, THREADS, 0, stream>>>(x, w, b, out, n);
}